// Gcn_13254269075791
// MI455X (gfx1250) — compile-verified
//
#include <hip/hip_runtime.h>
#include <math.h>

typedef float v2f __attribute__((ext_vector_type(2)));
typedef float v8f __attribute__((ext_vector_type(8)));

#define DIM     96
#define NGRAPH  64
#define NEG_SLOPE 0.2f

__device__ __forceinline__ float lrelu(float v) {
    return v > 0.f ? v : NEG_SLOPE * v;
}

// float atomic-max via signed-int max (v>=0) / unsigned-int min (v<0).
__device__ __forceinline__ void atomicMaxF(float* addr, float v) {
    if (v >= 0.f) atomicMax((int*)addr, __float_as_int(v));
    else          atomicMin((unsigned int*)addr, __float_as_uint(v));
}

// ---------------------------------------------------------------------------
// init: m = -inf, denom = 0, out = 0, pooled = 0, counts = 0
// ---------------------------------------------------------------------------
__global__ void k_init(float* __restrict__ m, float* __restrict__ denom,
                       float* __restrict__ out, float* __restrict__ pooled,
                       float* __restrict__ counts, int N) {
    int tid = blockIdx.x * blockDim.x + threadIdx.x;
    long long total = (long long)N * DIM;
    if (tid < total) out[tid] = 0.f;
    if (tid < N) { m[tid] = __int_as_float(0xff800000u); denom[tid] = 0.f; }
    if (tid < NGRAPH * DIM) pooled[tid] = 0.f;
    if (tid < NGRAPH)       counts[tid] = 0.f;
}

// ---------------------------------------------------------------------------
// h = x @ W_gat   (fp32 WMMA 16x16x4)
// One wave computes TWO 16-row tiles (32 rows x 96 cols): each B fragment
// pulled from LDS feeds 2 WMMAs, halving DS traffic / wait per matrix op.
// W_gat staged once per block into LDS as (K-pair, col) float2 pairs.
// A frag (16x4): lane l<16 holds (M=l, K=k..k+1), lane l+16 holds (M=l, K=k+2..k+3)
// B frag (4x16): VGPR j: lanes0-15 = K=j, lanes16-31 = K=j+2
// ---------------------------------------------------------------------------
__global__ void k_gemm_h(const float* __restrict__ x, const float* __restrict__ W,
                         float* __restrict__ h, int N) {
    __shared__ float2 Wl[(DIM / 2) * DIM];        // 48 K-pairs x 96 cols = 36 KB

    for (int i = threadIdx.x; i < (DIM / 2) * DIM; i += blockDim.x) {
        int mm = i / DIM, col = i - mm * DIM;
        Wl[i] = make_float2(W[(2 * mm) * DIM + col], W[(2 * mm + 1) * DIM + col]);
    }
    __syncthreads();

    int wave = (blockIdx.x * blockDim.x + threadIdx.x) >> 5;
    int lane = threadIdx.x & 31;
    int row0 = wave * 32;                         // two 16-row tiles
    if (row0 >= N) return;                        // wave-uniform

    int half = lane >> 4;
    int l16  = lane & 15;
    int ar0 = row0 + l16;      if (ar0 >= N) ar0 = N - 1;  // clamped: EXEC stays full
    int ar1 = row0 + 16 + l16; if (ar1 >= N) ar1 = N - 1;

    v8f acc0[6], acc1[6];
    #pragma unroll
    for (int t = 0; t < 6; ++t) {
        acc0[t] = (v8f){0,0,0,0,0,0,0,0};
        acc1[t] = (v8f){0,0,0,0,0,0,0,0};
    }

    const float* ap0 = x + (size_t)ar0 * DIM + 2 * half;
    const float* ap1 = x + (size_t)ar1 * DIM + 2 * half;

    for (int k = 0; k < DIM; k += 4) {
        v2f a0, a1;
        a0.x = ap0[k]; a0.y = ap0[k + 1];
        a1.x = ap1[k]; a1.y = ap1[k + 1];
        int mrow = (k >> 1) + half;               // K-pair row in LDS
        #pragma unroll
        for (int t = 0; t < 6; ++t) {
            float2 bp = Wl[mrow * DIM + t * 16 + l16];
            v2f b; b.x = bp.x; b.y = bp.y;
            acc0[t] = __builtin_amdgcn_wmma_f32_16x16x4_f32(
                false, a0, false, b, (short)0, acc0[t], false, false);
            acc1[t] = __builtin_amdgcn_wmma_f32_16x16x4_f32(
                false, a1, false, b, (short)0, acc1[t], false, false);
        }
    }

    // D layout: VGPR r -> (M = r + 8*half, N = l16) within tile
    float* hb0 = h + (size_t)(row0 + 8 * half) * DIM + l16;
    float* hb1 = hb0 + (size_t)16 * DIM;
    if (row0 + 32 <= N) {                         // wave-uniform fast path
        #pragma unroll
        for (int t = 0; t < 6; ++t)
            #pragma unroll
            for (int r = 0; r < 8; ++r) {
                hb0[(size_t)r * DIM + t * 16] = acc0[t][r];
                hb1[(size_t)r * DIM + t * 16] = acc1[t][r];
            }
    } else {                                      // tail: guarded stores
        #pragma unroll
        for (int t = 0; t < 6; ++t)
            #pragma unroll
            for (int r = 0; r < 8; ++r) {
                if (row0 + 8 * half + r < N)
                    hb0[(size_t)r * DIM + t * 16] = acc0[t][r];
                if (row0 + 16 + 8 * half + r < N)
                    hb1[(size_t)r * DIM + t * 16] = acc1[t][r];
            }
    }
}

// ---------------------------------------------------------------------------
// a_s[i] = h[i,:]·att_src ; a_d[i] = h[i,:]·att_dst  (one wave per node)
// ---------------------------------------------------------------------------
__global__ void k_att(const float* __restrict__ h, const float* __restrict__ att_s,
                      const float* __restrict__ att_d, float* __restrict__ a_s,
                      float* __restrict__ a_d, int N) {
    int wave = (blockIdx.x * blockDim.x + threadIdx.x) >> 5;
    int lane = threadIdx.x & 31;
    if (wave >= N) return;
    float sa = 0.f, sd = 0.f;
    #pragma unroll
    for (int c = 0; c < 3; ++c) {
        int d = lane + 32 * c;
        float hv = h[(size_t)wave * DIM + d];
        sa += hv * att_s[d];
        sd += hv * att_d[d];
    }
    #pragma unroll
    for (int off = 16; off >= 1; off >>= 1) {
        sa += __shfl_xor(sa, off, 32);
        sd += __shfl_xor(sd, off, 32);
    }
    if (lane == 0) { a_s[wave] = sa; a_d[wave] = sd; }
}

// ---------------------------------------------------------------------------
// pass 1: segment max of leaky_relu(a_s[src]+a_d[dst]) over dst
// ---------------------------------------------------------------------------
__global__ void k_edge_max(const int* __restrict__ ei, const float* __restrict__ a_s,
                           const float* __restrict__ a_d, float* __restrict__ m,
                           int E, int Etot) {
    int e = blockIdx.x * blockDim.x + threadIdx.x;
    if (e >= Etot) return;
    int s, d;
    if (e < E) { s = ei[e]; d = ei[E + e]; } else { s = d = e - E; }
    atomicMaxF(&m[d], lrelu(a_s[s] + a_d[d]));
}

// ---------------------------------------------------------------------------
// pass 2: p = exp(e - m[dst]); denom[dst] += p
// ---------------------------------------------------------------------------
__global__ void k_edge_sum(const int* __restrict__ ei, const float* __restrict__ a_s,
                           const float* __restrict__ a_d, const float* __restrict__ m,
                           float* __restrict__ p, float* __restrict__ denom,
                           int E, int Etot) {
    int e = blockIdx.x * blockDim.x + threadIdx.x;
    if (e >= Etot) return;
    int s, d;
    if (e < E) { s = ei[e]; d = ei[E + e]; } else { s = d = e - E; }
    float pe = __expf(lrelu(a_s[s] + a_d[d]) - m[d]);
    p[e] = pe;
    atomicAdd(&denom[d], pe);
}

// ---------------------------------------------------------------------------
// pass 3: out[dst] += (p/denom[dst]) * h[src]   (one wave per edge, 3 dims/lane)
// ---------------------------------------------------------------------------
__global__ void k_edge_agg(const int* __restrict__ ei, const float* __restrict__ h,
                           const float* __restrict__ p, const float* __restrict__ denom,
                           float* __restrict__ out, int E, int Etot) {
    int wave = (blockIdx.x * blockDim.x + threadIdx.x) >> 5;
    int lane = threadIdx.x & 31;
    if (wave >= Etot) return;
    int s, d;
    if (wave < E) { s = ei[wave]; d = ei[E + wave]; } else { s = d = wave - E; }
    float coef = p[wave] / denom[d];
    #pragma unroll
    for (int c = 0; c < 3; ++c) {
        int dd = lane + 32 * c;
        atomicAdd(&out[(size_t)d * DIM + dd], coef * h[(size_t)s * DIM + dd]);
    }
}

// ---------------------------------------------------------------------------
// relu(out + bias) mean-pool by graph (one wave per node)
// ---------------------------------------------------------------------------
__global__ void k_pool(const float* __restrict__ out, const float* __restrict__ bias,
                       const int* __restrict__ batch, float* __restrict__ pooled,
                       float* __restrict__ counts, int N) {
    int wave = (blockIdx.x * blockDim.x + threadIdx.x) >> 5;
    int lane = threadIdx.x & 31;
    if (wave >= N) return;
    int g = batch[wave];
    #pragma unroll
    for (int c = 0; c < 3; ++c) {
        int d = lane + 32 * c;
        float v = out[(size_t)wave * DIM + d] + bias[d];
        v = v > 0.f ? v : 0.f;
        atomicAdd(&pooled[g * DIM + d], v);
    }
    if (lane == 0) atomicAdd(&counts[g], 1.0f);
}

__global__ void k_norm(float* __restrict__ pooled, const float* __restrict__ counts) {
    int tid = blockIdx.x * blockDim.x + threadIdx.x;
    if (tid >= NGRAPH * DIM) return;
    pooled[tid] /= fmaxf(counts[tid / DIM], 1.0f);
}

// ---------------------------------------------------------------------------
// result = pooled(64x96) @ W_lin + b_lin   (WMMA, 4x6 = 24 tiles = 24 waves)
// ---------------------------------------------------------------------------
__global__ void k_gemm_final(const float* __restrict__ pooled, const float* __restrict__ W,
                             const float* __restrict__ b, float* __restrict__ res) {
    int wave = (blockIdx.x * blockDim.x + threadIdx.x) >> 5;
    int lane = threadIdx.x & 31;
    if (wave >= 24) return;
    int rt = wave / 6, ct = wave % 6;
    int row0 = rt * 16, n0 = ct * 16;
    int half = lane >> 4, l16 = lane & 15;

    v8f acc = (v8f){0,0,0,0,0,0,0,0};
    for (int k = 0; k < DIM; k += 4) {
        v2f a;
        const float* ap = pooled + (size_t)(row0 + l16) * DIM + k + 2 * half;
        a.x = ap[0]; a.y = ap[1];
        v2f bf;
        int col = n0 + l16;
        bf.x = W[(k + 2 * half    ) * DIM + col];
        bf.y = W[(k + 2 * half + 1) * DIM + col];
        acc = __builtin_amdgcn_wmma_f32_16x16x4_f32(
            false, a, false, bf, (short)0, acc, false, false);
    }
    float bias = b[n0 + l16];
    #pragma unroll
    for (int r = 0; r < 8; ++r) {
        int row = row0 + r + 8 * half;
        res[(size_t)row * DIM + n0 + l16] = acc[r] + bias;
    }
}

// ---------------------------------------------------------------------------
extern "C" void kernel_launch(void* const* d_in, const int* in_sizes, int n_in,
                              void* d_out, int out_size, void* d_ws, size_t ws_size,
                              hipStream_t stream) {
    const float* x        = (const float*)d_in[0];
    const int*   ei       = (const int*)  d_in[1];   // [2,E]: src then dst
    const int*   batch    = (const int*)  d_in[3];
    const float* W_gat    = (const float*)d_in[4];
    const float* att_src  = (const float*)d_in[5];
    const float* att_dst  = (const float*)d_in[6];
    const float* bias_gat = (const float*)d_in[7];
    const float* W_lin    = (const float*)d_in[8];
    const float* b_lin    = (const float*)d_in[9];
    float* res = (float*)d_out;

    int N = in_sizes[0] / DIM;
    int E = in_sizes[1] / 2;
    int Etot = E + N;

    // workspace layout (fp32)
    float* h      = (float*)d_ws;
    float* outbuf = h + (size_t)N * DIM;
    float* a_s    = outbuf + (size_t)N * DIM;
    float* a_d    = a_s + N;
    float* m      = a_d + N;
    float* denom  = m + N;
    float* p      = denom + N;
    float* pooled = p + Etot;
    float* counts = pooled + NGRAPH * DIM;

    const int B = 256;
    long long initN = (long long)N * DIM;
    k_init<<<(int)((initN + B - 1) / B), B, 0, stream>>>(m, denom, outbuf, pooled, counts, N);

    int waves32 = (N + 31) / 32;                  // one wave per 32 rows
    k_gemm_h<<<(waves32 + 7) / 8, B, 0, stream>>>(x, W_gat, h, N);

    k_att<<<(N + 7) / 8, B, 0, stream>>>(h, att_src, att_dst, a_s, a_d, N);

    k_edge_max<<<(Etot + B - 1) / B, B, 0, stream>>>(ei, a_s, a_d, m, E, Etot);
    k_edge_sum<<<(Etot + B - 1) / B, B, 0, stream>>>(ei, a_s, a_d, m, p, denom, E, Etot);
    k_edge_agg<<<(Etot + 7) / 8, B, 0, stream>>>(ei, h, p, denom, outbuf, E, Etot);

    k_pool<<<(N + 7) / 8, B, 0, stream>>>(outbuf, bias_gat, batch, pooled, counts, N);
    k_norm<<<(NGRAPH * DIM + B - 1) / B, B, 0, stream>>>(pooled, counts);

    k_gemm_final<<<3, B, 0, stream>>>(pooled, W_lin, b_lin, res);
}